// NSF_M_all_uncond_84035330113543
// MI455X (gfx1250) — compile-verified
//
#include <hip/hip_runtime.h>
#include <math.h>

#define KBINS 5
#define NFL 4
#define BB 3.0f

// d_ws float layout:
//   [0 .. 160)   per-flow bin records: flow f at f*40 floats, bin k at +k*8:
//                { cw_k, 1/w_k, ch_k, h_k,  delta_k, d_l, d_r, s=(d_l+d_r-2*delta) }
//   [160 .. 176) interior knots: flow f at 160+f*4 : { cumw1, cumw2, cumw3, cumw4 }
#define TBL_FLOATS 176

typedef unsigned int u32x4 __attribute__((ext_vector_type(4)));
typedef int          i32x8 __attribute__((ext_vector_type(8)));
typedef int          i32x4 __attribute__((ext_vector_type(4)));

// ---------------- setup: tiny, one wave, full-precision libm ----------------
__global__ void nsf_setup(const float* __restrict__ layers, float* __restrict__ ws) {
    int f = threadIdx.x;
    if (f >= NFL) return;
    const float* p = layers + f * (3 * KBINS - 1);
    float W[KBINS], H[KBINS], Dm[KBINS + 1];

    float m = p[0];
    for (int k = 1; k < KBINS; ++k) m = fmaxf(m, p[k]);
    float s = 0.f;
    for (int k = 0; k < KBINS; ++k) { W[k] = expf(p[k] - m); s += W[k]; }
    float sc = (2.0f * BB) / s;
    for (int k = 0; k < KBINS; ++k) W[k] *= sc;

    m = p[KBINS];
    for (int k = 1; k < KBINS; ++k) m = fmaxf(m, p[KBINS + k]);
    s = 0.f;
    for (int k = 0; k < KBINS; ++k) { H[k] = expf(p[KBINS + k] - m); s += H[k]; }
    sc = (2.0f * BB) / s;
    for (int k = 0; k < KBINS; ++k) H[k] *= sc;

    Dm[0] = 1.0f; Dm[KBINS] = 1.0f;
    for (int k = 0; k < KBINS - 1; ++k)
        Dm[k + 1] = 2.0f / (1.0f + expf(-p[2 * KBINS + k]));

    float cw = -BB, ch = -BB;
    float* bp = ws + f * 40;
    float* kn = ws + 160 + f * 4;
    for (int k = 0; k < KBINS; ++k) {
        float w = W[k], h = H[k];
        float winv  = 1.0f / w;
        float delta = h / w;
        float dl = Dm[k], dr = Dm[k + 1];
        bp[k * 8 + 0] = cw;    bp[k * 8 + 1] = winv;
        bp[k * 8 + 2] = ch;    bp[k * 8 + 3] = h;
        bp[k * 8 + 4] = delta; bp[k * 8 + 5] = dl;
        bp[k * 8 + 6] = dr;    bp[k * 8 + 7] = dl + dr - 2.0f * delta;
        cw += w; ch += h;
        if (k < 4) kn[k] = cw;          // cumw[1..4]
    }
}

// ---------------- per-element spline chain ----------------
__device__ __forceinline__ void rqs_elem(float x, const float* __restrict__ tbl,
                                         const float (&kn)[NFL][4],
                                         float mu, float c0, float kq,
                                         float& logp, float& ldet, float& zout) {
    float z = fminf(fmaxf(x, -BB), BB);             // clamp only needed for flow 0
    bool inside = (x >= -BB) && (x <= BB);
    float lacc = 0.0f;                              // accumulated log2|J|
#pragma unroll
    for (int f = 0; f < NFL; ++f) {
        float xc = z;                               // in [-B,B] by construction
        int idx = (xc >= kn[f][0]) + (xc >= kn[f][1]) +
                  (xc >= kn[f][2]) + (xc >= kn[f][3]);
        const float4* bp = reinterpret_cast<const float4*>(tbl + f * 40 + idx * 8);
        float4 A = bp[0];                           // cw, 1/w, ch, h
        float4 Bv = bp[1];                          // delta, dl, dr, s
        float theta = (xc - A.x) * A.y;
        float omt   = 1.0f - theta;
        float tomt  = theta * omt;
        float t2    = theta * theta;
        float denom = fmaf(Bv.w, tomt, Bv.x);       // d + s*th(1-th)   (>0)
        float r     = __builtin_amdgcn_rcpf(denom);
        float num   = A.w * fmaf(Bv.y, tomt, Bv.x * t2);
        z = fmaf(num, r, A.z);                      // ch + h*(..)/denom
        float inner = fmaf(Bv.z, t2,
                      fmaf(Bv.y, omt * omt, (Bv.x + Bv.x) * tomt));
        float dnum  = (Bv.x * Bv.x) * inner;        // d^2 * (...)      (>0)
        lacc += __builtin_amdgcn_logf(dnum * (r * r));   // log2(dnum/denom^2)
    }
    zout = inside ? z : x;
    ldet = inside ? lacc * 0.69314718055994530942f : 0.0f;
    float dx = zout - mu;
    logp = fmaf(dx * dx, kq, c0);
}

// ---------------- main streaming kernel ----------------
__global__ __launch_bounds__(256) void nsf_main(const float* __restrict__ x,
                                                const float* __restrict__ ip,
                                                const float* __restrict__ ws,
                                                float* __restrict__ out, int n) {
    __shared__ __align__(16) float tbl[256];        // 176 used + headroom

#if defined(__gfx1250__) && __has_builtin(__builtin_amdgcn_tensor_load_to_lds)
    // TDM broadcast of the 704B table into LDS (wave 0 issues; EXEC ignored).
    // amdgpu-toolchain (clang-23) 6-arg form:
    //   (uint32x4 g0, int32x8 g1, int32x4 g2, int32x4 g3, int32x8 g4, i32 cpol)
    if (threadIdx.x < 32) {
        unsigned long long ga = (unsigned long long)ws;
        unsigned lds_off = (unsigned)(unsigned long long)(&tbl[0]); // flat low32 = LDS offset
        u32x4 g0 = { 1u,                                  // count=1 (valid descriptor)
                     lds_off,                             // lds_addr [63:32]
                     (unsigned)(ga & 0xffffffffu),        // global_addr lo [95:64]
                     (unsigned)((ga >> 32) & 0x1ffffffu) | (2u << 30) }; // ga hi + type=2
        i32x8 g1 = { 0x20000,                             // data_size=4B, mask=0
                     (int)(TBL_FLOATS << 16),             // tensor_dim0 lo16 @[63:48]
                     0x10000,                             // tensor_dim1=1 lo16 @[95:80]
                     (int)(TBL_FLOATS << 16),             // tile_dim0 @[127:112]
                     1,                                   // tile_dim1=1, tile_dim2=0
                     TBL_FLOATS,                          // tensor_dim0_stride lo32
                     (int)(TBL_FLOATS << 16),             // tensor_dim1_stride lo16
                     0 };
        i32x4 gz4 = { 0, 0, 0, 0 };
        i32x8 gz8 = { 0, 0, 0, 0, 0, 0, 0, 0 };
        __builtin_amdgcn_tensor_load_to_lds(g0, g1, gz4, gz4, gz8, 0);
        __builtin_amdgcn_s_wait_tensorcnt((short)0);
    }
    __syncthreads();   // TDM data landed before the verifying fill below
#endif
    // Guaranteed-correct cooperative fill (overwrites TDM result with same data).
    for (int i = threadIdx.x; i < TBL_FLOATS; i += blockDim.x) tbl[i] = ws[i];
    __syncthreads();

    // Uniform Gaussian constants (scalar-cached loads; 1 exp per thread).
    float mu = ip[0];
    float lv = ip[1];
    float invvar = expf(-lv);
    float c0 = -0.91893853320467274178f - 0.5f * lv;   // -0.5*log(2*pi) - 0.5*logvar
    float kq = -0.5f * invvar;

    // Interior knots into registers (16 VGPRs).
    float kn[NFL][4];
#pragma unroll
    for (int f = 0; f < NFL; ++f)
#pragma unroll
        for (int t = 0; t < 4; ++t) kn[f][t] = tbl[160 + f * 4 + t];

    const int vecN = n >> 2;
    const float4* __restrict__ xv = reinterpret_cast<const float4*>(x);
    float4* o_lp = reinterpret_cast<float4*>(out);
    float4* o_ld = reinterpret_cast<float4*>(out + n);
    float4* o_z  = reinterpret_cast<float4*>(out + 2 * n);

    const int stride = gridDim.x * blockDim.x;
    for (int i = blockIdx.x * blockDim.x + threadIdx.x; i < vecN; i += stride) {
        float4 xx = xv[i];                 // global_load_b128
        float4 lp, ld, zz;
        rqs_elem(xx.x, tbl, kn, mu, c0, kq, lp.x, ld.x, zz.x);
        rqs_elem(xx.y, tbl, kn, mu, c0, kq, lp.y, ld.y, zz.y);
        rqs_elem(xx.z, tbl, kn, mu, c0, kq, lp.z, ld.z, zz.z);
        rqs_elem(xx.w, tbl, kn, mu, c0, kq, lp.w, ld.w, zz.w);
        o_lp[i] = lp;                      // global_store_b128 x3
        o_ld[i] = ld;
        o_z[i]  = zz;
    }
    // Scalar tail for n % 4 != 0 (not hit for N=8388608, kept for robustness).
    for (int i = (vecN << 2) + blockIdx.x * blockDim.x + threadIdx.x; i < n; i += stride) {
        float lp, ld, zz;
        rqs_elem(x[i], tbl, kn, mu, c0, kq, lp, ld, zz);
        out[i] = lp; out[n + i] = ld; out[2 * n + i] = zz;
    }
}

extern "C" void kernel_launch(void* const* d_in, const int* in_sizes, int n_in,
                              void* d_out, int out_size, void* d_ws, size_t ws_size,
                              hipStream_t stream) {
    const float* x      = (const float*)d_in[0];
    const float* layers = (const float*)d_in[1];
    const float* ip     = (const float*)d_in[2];
    float* out = (float*)d_out;
    float* ws  = (float*)d_ws;
    int n = in_sizes[0];

    nsf_setup<<<1, 32, 0, stream>>>(layers, ws);

    int threads = 256;
    int vecN = n >> 2;
    int blocks = 4096;                                  // spread across WGPs for HBM
    int need = (vecN + threads - 1) / threads;
    if (need < 1) need = 1;
    if (blocks > need) blocks = need;
    nsf_main<<<blocks, threads, 0, stream>>>(x, ip, ws, out, n);
}